// RGCN_82454782148689
// MI455X (gfx1250) — compile-verified
//
#include <hip/hip_runtime.h>
#include <hip/hip_bf16.h>

#define NUM_TX   100000
#define NUM_USER 30000
#define NUM_DEV  10000
#define NNODES   (NUM_TX + NUM_USER + NUM_DEV)   // 140000, divisible by 16
#define IN_C     128
#define HID      256
#define NUM_RELS 4
#define NEDGE    1000000
#define BN_EPS   1e-5f

typedef float v2f __attribute__((ext_vector_type(2)));
typedef float v8f __attribute__((ext_vector_type(8)));

// ---------------------------------------------------------------------------
// Edge scatter: AGG[r][dst] += X[src], CNT[r][dst] += 1  (per-relation mean num/den)
// One float4 chunk per thread; C/4 threads per edge.
// ---------------------------------------------------------------------------
__global__ void rgcn_edge_scatter(const float* __restrict__ X,
                                  const int* __restrict__ src,
                                  const int* __restrict__ dst,
                                  const int* __restrict__ et,
                                  float* __restrict__ AGG,
                                  float* __restrict__ CNT,
                                  int C, int log_cpe) {
    const long long t = (long long)blockIdx.x * blockDim.x + threadIdx.x;
    const long long total = (long long)NEDGE << log_cpe;
    if (t >= total) return;
    const int e  = (int)(t >> log_cpe);
    const int ch = ((int)t) & ((1 << log_cpe) - 1);
    const int r = et[e];
    const int s = src[e];
    const int d = dst[e];
    const float4 v = *(const float4*)(X + (size_t)s * C + ch * 4);
    float* o = AGG + ((size_t)r * NNODES + d) * C + ch * 4;
    atomicAdd(o + 0, v.x);
    atomicAdd(o + 1, v.y);
    atomicAdd(o + 2, v.z);
    atomicAdd(o + 3, v.w);
    if (ch == 0) atomicAdd(CNT + (size_t)r * NNODES + d, 1.0f);
}

__global__ void rgcn_inv_counts(const float* __restrict__ CNT,
                                float* __restrict__ INV) {
    const int i = blockIdx.x * blockDim.x + threadIdx.x;
    if (i < NUM_RELS * NNODES) INV[i] = 1.0f / fmaxf(CNT[i], 1.0f);
}

// ---------------------------------------------------------------------------
// Fused RGCN dense stage (fp32 WMMA):
//   H[m, n] = bias[n] + X[m,:] @ rootW[:,n] + sum_r (AGG[r][m,:] * INV[r][m]) @ relW[r][:,n]
// One wave computes one 16x16 output tile via chained V_WMMA_F32_16X16X4_F32.
// 16 waves / block cover the full HID=256 for one 16-node tile.
// Lane layouts per CDNA5 ISA 7.12.2:
//   A (16x4 f32):  M = lane%16, K(frag elem v) = v + 2*(lane/16)
//   B (4x16 f32):  N = lane%16, K(frag elem v) = v + 2*(lane/16)
//   C/D (16x16):   N = lane%16, M(vgpr i)      = i + 8*(lane/16)
// ---------------------------------------------------------------------------
__global__ __launch_bounds__(512) void rgcn_gemm_wmma(
    const float* __restrict__ X,      // [NNODES, K]
    const float* __restrict__ AGG,    // [NUM_RELS, NNODES, K]
    const float* __restrict__ INV,    // [NUM_RELS, NNODES]
    const float* __restrict__ rootW,  // [K, HID]
    const float* __restrict__ relW,   // [NUM_RELS, K, HID]
    const float* __restrict__ bias,   // [HID]
    float* __restrict__ H,            // [NNODES, HID]
    int K) {
    const int lane  = threadIdx.x & 31;
    const int wave  = threadIdx.x >> 5;          // 0..15 -> column tile
    const int mbase = blockIdx.x * 16;
    const int nloc  = lane & 15;
    const int ncol  = wave * 16 + nloc;
    const int khalf = lane >> 4;                 // 0 or 1
    const int m     = mbase + nloc;              // A row handled by this lane
    const size_t rowX = (size_t)m * K;

    v8f acc = {0.f, 0.f, 0.f, 0.f, 0.f, 0.f, 0.f, 0.f};

    // Root path: X @ rootW
    for (int kb = 0; kb < K; kb += 4) {
        const int k0 = kb + 2 * khalf;
        v2f a, b;
        a.x = X[rowX + k0];
        a.y = X[rowX + k0 + 1];
        b.x = rootW[(size_t)k0 * HID + ncol];
        b.y = rootW[(size_t)(k0 + 1) * HID + ncol];
        acc = __builtin_amdgcn_wmma_f32_16x16x4_f32(false, a, false, b,
                                                    (short)0, acc, false, false);
    }
    // Relation paths: (AGG[r]/cnt) @ relW[r]  (mean folded into A-fragment load)
    for (int r = 0; r < NUM_RELS; ++r) {
        const float inv = INV[(size_t)r * NNODES + m];
        const float* __restrict__ Ar = AGG + ((size_t)r * NNODES + m) * K;
        const float* __restrict__ Wr = relW + (size_t)r * K * HID;
        for (int kb = 0; kb < K; kb += 4) {
            const int k0 = kb + 2 * khalf;
            v2f a, b;
            a.x = Ar[k0] * inv;
            a.y = Ar[k0 + 1] * inv;
            b.x = Wr[(size_t)k0 * HID + ncol];
            b.y = Wr[(size_t)(k0 + 1) * HID + ncol];
            acc = __builtin_amdgcn_wmma_f32_16x16x4_f32(false, a, false, b,
                                                        (short)0, acc, false, false);
        }
    }

    const float bv = bias[ncol];
#pragma unroll
    for (int i = 0; i < 8; ++i) {
        const int row = mbase + i + 8 * khalf;
        H[(size_t)row * HID + ncol] = acc[i] + bv;
    }
}

// ---------------------------------------------------------------------------
// BatchNorm (training-mode, biased stats over node dim)
// ---------------------------------------------------------------------------
__global__ __launch_bounds__(256) void bn_stats(const float* __restrict__ H,
                                                float* __restrict__ sums,
                                                float* __restrict__ sqs) {
    const int c = threadIdx.x;                    // column 0..255
    const int row0 = blockIdx.x * 256;
    const int rend = min(row0 + 256, NNODES);
    float s = 0.f, s2 = 0.f;
    for (int r = row0; r < rend; ++r) {
        const float v = H[(size_t)r * HID + c];
        s += v;
        s2 += v * v;
    }
    atomicAdd(&sums[c], s);
    atomicAdd(&sqs[c], s2);
}

__global__ void bn_finalize(const float* __restrict__ sums,
                            const float* __restrict__ sqs,
                            const float* __restrict__ gamma,
                            const float* __restrict__ beta,
                            float* __restrict__ scale,
                            float* __restrict__ shift) {
    const int c = threadIdx.x;
    if (c < HID) {
        const float mean = sums[c] * (1.0f / (float)NNODES);
        const float var  = sqs[c] * (1.0f / (float)NNODES) - mean * mean;
        const float sc   = gamma[c] * rsqrtf(var + BN_EPS);
        scale[c] = sc;
        shift[c] = beta[c] - mean * sc;
    }
}

__global__ void bn_apply_relu(float* __restrict__ H,
                              const float* __restrict__ scale,
                              const float* __restrict__ shift) {
    const long long t = (long long)blockIdx.x * blockDim.x + threadIdx.x;
    if (t >= (long long)NNODES * (HID / 4)) return;
    const int c4 = (((int)t) & (HID / 4 - 1)) * 4;
    float4 v = ((float4*)H)[t];
    v.x = fmaxf(v.x * scale[c4 + 0] + shift[c4 + 0], 0.f);
    v.y = fmaxf(v.y * scale[c4 + 1] + shift[c4 + 1], 0.f);
    v.z = fmaxf(v.z * scale[c4 + 2] + shift[c4 + 2], 0.f);
    v.w = fmaxf(v.w * scale[c4 + 3] + shift[c4 + 3], 0.f);
    ((float4*)H)[t] = v;
}

// ---------------------------------------------------------------------------
// Fused BN2-apply + ReLU + classifier over the first NUM_TX rows.
// One wave32 per row; shuffle reduction.
// ---------------------------------------------------------------------------
__global__ __launch_bounds__(256) void bn_relu_classify(
    const float* __restrict__ H2, const float* __restrict__ scale,
    const float* __restrict__ shift, const float* __restrict__ clsW, // [HID,2]
    const float* __restrict__ clsB, float* __restrict__ out) {
    const int lane = threadIdx.x & 31;
    const int wave = threadIdx.x >> 5;
    const int row  = blockIdx.x * 8 + wave;
    if (row >= NUM_TX) return;
    float a0 = 0.f, a1 = 0.f;
    const float* __restrict__ hr = H2 + (size_t)row * HID;
    for (int k = lane; k < HID; k += 32) {
        const float h = fmaxf(hr[k] * scale[k] + shift[k], 0.f);
        a0 += h * clsW[k * 2 + 0];
        a1 += h * clsW[k * 2 + 1];
    }
#pragma unroll
    for (int off = 16; off > 0; off >>= 1) {
        a0 += __shfl_down(a0, off, 32);
        a1 += __shfl_down(a1, off, 32);
    }
    if (lane == 0) {
        out[row * 2 + 0] = a0 + clsB[0];
        out[row * 2 + 1] = a1 + clsB[1];
    }
}

// ---------------------------------------------------------------------------
extern "C" void kernel_launch(void* const* d_in, const int* in_sizes, int n_in,
                              void* d_out, int out_size, void* d_ws, size_t ws_size,
                              hipStream_t stream) {
    const float* x_tx    = (const float*)d_in[0];
    const float* user_e  = (const float*)d_in[1];
    const float* dev_e   = (const float*)d_in[2];
    const float* rel_w1  = (const float*)d_in[3];
    const float* root_w1 = (const float*)d_in[4];
    const float* bias1   = (const float*)d_in[5];
    const float* rel_w2  = (const float*)d_in[6];
    const float* root_w2 = (const float*)d_in[7];
    const float* bias2   = (const float*)d_in[8];
    const float* g1      = (const float*)d_in[9];
    const float* b1      = (const float*)d_in[10];
    const float* g2      = (const float*)d_in[11];
    const float* b2      = (const float*)d_in[12];
    const float* cls_w   = (const float*)d_in[13];
    const float* cls_b   = (const float*)d_in[14];
    const int*   eidx    = (const int*)d_in[15];
    const int*   etype   = (const int*)d_in[16];
    const int*   srcp    = eidx;
    const int*   dstp    = eidx + NEDGE;

    // Workspace layout (floats)
    float* ws = (float*)d_ws;
    size_t o = 0;
    float* X   = ws + o; o += (size_t)NNODES * IN_C;            // 140000*128
    float* H1  = ws + o; o += (size_t)NNODES * HID;
    float* H2  = ws + o; o += (size_t)NNODES * HID;
    float* AGG = ws + o; o += (size_t)NUM_RELS * NNODES * HID;  // reused, layer1 uses /2
    float* CNT = ws + o; o += (size_t)NUM_RELS * NNODES;
    float* INV = ws + o; o += (size_t)NUM_RELS * NNODES;
    float* sums  = ws + o; o += HID;   // sums & sqs contiguous -> one memset
    float* sqs   = ws + o; o += HID;
    float* scale = ws + o; o += HID;
    float* shift = ws + o; o += HID;

    // X = concat(x_tx, user_emb, device_emb)
    hipMemcpyAsync(X, x_tx, (size_t)NUM_TX * IN_C * 4, hipMemcpyDeviceToDevice, stream);
    hipMemcpyAsync(X + (size_t)NUM_TX * IN_C, user_e, (size_t)NUM_USER * IN_C * 4,
                   hipMemcpyDeviceToDevice, stream);
    hipMemcpyAsync(X + (size_t)(NUM_TX + NUM_USER) * IN_C, dev_e,
                   (size_t)NUM_DEV * IN_C * 4, hipMemcpyDeviceToDevice, stream);

    const int inv_n = NUM_RELS * NNODES;

    // ---------------- Layer 1 (K = 128) ----------------
    hipMemsetAsync(AGG, 0, (size_t)NUM_RELS * NNODES * IN_C * 4, stream);
    hipMemsetAsync(CNT, 0, (size_t)inv_n * 4, stream);
    {
        const long long tot = (long long)NEDGE * (IN_C / 4);
        rgcn_edge_scatter<<<(int)((tot + 255) / 256), 256, 0, stream>>>(
            X, srcp, dstp, etype, AGG, CNT, IN_C, 5);
    }
    rgcn_inv_counts<<<(inv_n + 255) / 256, 256, 0, stream>>>(CNT, INV);
    rgcn_gemm_wmma<<<NNODES / 16, 512, 0, stream>>>(X, AGG, INV, root_w1, rel_w1,
                                                    bias1, H1, IN_C);
    hipMemsetAsync(sums, 0, 2 * HID * 4, stream);
    bn_stats<<<(NNODES + 255) / 256, 256, 0, stream>>>(H1, sums, sqs);
    bn_finalize<<<1, 256, 0, stream>>>(sums, sqs, g1, b1, scale, shift);
    {
        const long long tot = (long long)NNODES * (HID / 4);
        bn_apply_relu<<<(int)((tot + 255) / 256), 256, 0, stream>>>(H1, scale, shift);
    }

    // ---------------- Layer 2 (K = 256) ----------------
    hipMemsetAsync(AGG, 0, (size_t)NUM_RELS * NNODES * HID * 4, stream);
    hipMemsetAsync(CNT, 0, (size_t)inv_n * 4, stream);
    {
        const long long tot = (long long)NEDGE * (HID / 4);
        rgcn_edge_scatter<<<(int)((tot + 255) / 256), 256, 0, stream>>>(
            H1, srcp, dstp, etype, AGG, CNT, HID, 6);
    }
    rgcn_inv_counts<<<(inv_n + 255) / 256, 256, 0, stream>>>(CNT, INV);
    rgcn_gemm_wmma<<<NNODES / 16, 512, 0, stream>>>(H1, AGG, INV, root_w2, rel_w2,
                                                    bias2, H2, HID);
    hipMemsetAsync(sums, 0, 2 * HID * 4, stream);
    bn_stats<<<(NNODES + 255) / 256, 256, 0, stream>>>(H2, sums, sqs);
    bn_finalize<<<1, 256, 0, stream>>>(sums, sqs, g2, b2, scale, shift);

    // Fused BN2-apply + ReLU + classifier (transaction rows only)
    bn_relu_classify<<<NUM_TX / 8, 256, 0, stream>>>(H2, scale, shift, cls_w, cls_b,
                                                     (float*)d_out);
}